// AttentionModel_16140487098588
// MI455X (gfx1250) — compile-verified
//
#include <hip/hip_runtime.h>

// ---------------------------------------------------------------------------
// AttentionModel on MI455X (gfx1250), wave32.
//   kernel 1: h = emb[x];  Q|K|V = h @ W{q,k,v}^T  (fused gather + f32 WMMA GEMM)
//   kernel 2: flash attention with TDM (tensor_load_to_lds) K/V staging,
//             double-buffered on TENSORcnt, f32 WMMA for QK^T and P.V.
// Scores scaled by 1/sqrt(EMBED_DIM) = 1/16 (folded into Q fragments).
// ---------------------------------------------------------------------------

typedef float v2f __attribute__((ext_vector_type(2)));
typedef float v8f __attribute__((ext_vector_type(8)));
typedef unsigned int u32x4 __attribute__((ext_vector_type(4)));
typedef int i32x4 __attribute__((ext_vector_type(4)));
typedef int i32x8 __attribute__((ext_vector_type(8)));

#define B_  4
#define S_  4096
#define E_  256
#define D_  64
#define NB_ (S_ / 16)        // key blocks per batch
#define LSTRIDE 66           // LDS row stride in dwords (64 + 2 pad, conflict-free)
#define TILEDW (16 * LSTRIDE)

// 8-arg pattern: (neg_a, A, neg_b, B, c_mod, C, reuse_a, reuse_b)
__device__ __forceinline__ v8f wmma_f32(v2f a, v2f b, v8f c) {
  return __builtin_amdgcn_wmma_f32_16x16x4_f32(false, a, false, b, (short)0, c,
                                               false, false);
}

// ---------------------------------------------------------------------------
// TDM: DMA one 16x64 f32 tile (row stride 64 in memory) into LDS at lds_off,
// inserting 2 pad dwords after every 64 dwords -> LDS row stride 66.
// D# per CDNA5 ISA ch.8: group0 = {count|flags, lds_addr, global_addr, type},
// group1 = {mask/size/pad, dims/strides}, groups 2/3 unused (2-D tile).
// ---------------------------------------------------------------------------
__device__ __forceinline__ void tdm_load_tile16x64(unsigned lds_off,
                                                   const float* gptr) {
  const unsigned long long ga = (unsigned long long)(uintptr_t)gptr;
  u32x4 g0;
  g0.x = 1u;                                   // count=1 valid descriptor
  g0.y = lds_off;                              // lds_addr (bytes)
  g0.z = (unsigned)(ga & 0xffffffffu);         // global_addr[31:0]
  g0.w = (unsigned)((ga >> 32) & 0x01ffffffu)  // global_addr[56:32]
         | (2u << 30);                         // type = 2 ("image")
  i32x8 g1;
  g1[0] = (2 << 16)                            // data_size = 4 bytes
        | (1 << 20)                            // pad_enable
        | (5 << 22)                            // pad_interval: 64 dwords
        | (1 << 25);                           // pad_amount:   2 dwords
  g1[1] = (D_ & 0xffff) << 16;                 // tensor_dim0[15:0] = 64
  g1[2] = ((D_ >> 16) & 0xffff)                // tensor_dim0[31:16] = 0
        | ((65536 & 0xffff) << 16);            // tensor_dim1[15:0] = 0
  g1[3] = ((65536 >> 16) & 0xffff)             // tensor_dim1[31:16] -> 65536 rows
        | (D_ << 16);                          // tile_dim0 = 64
  g1[4] = 16;                                  // tile_dim1 = 16, tile_dim2 = 0
  g1[5] = D_;                                  // tensor_dim0_stride[31:0] = 64
  g1[6] = 0;                                   // stride hi / dim1_stride lo
  g1[7] = 0;
  i32x4 zz = {0, 0, 0, 0};                     // 2-D tile: groups 2/3 unused
#if defined(__clang_major__) && (__clang_major__ >= 23)
  i32x8 z8 = {0, 0, 0, 0, 0, 0, 0, 0};
  __builtin_amdgcn_tensor_load_to_lds(g0, g1, zz, zz, z8, 0);
#else
  __builtin_amdgcn_tensor_load_to_lds(g0, g1, zz, zz, 0);
#endif
}

// ---------------------------------------------------------------------------
// Kernel 1: embedding gather fused with QKV projection.
// Grid: (BS/16, 12); one wave per 16x16 tile, 64 WMMAs over K=256.
// ---------------------------------------------------------------------------
__global__ __launch_bounds__(32) void qkv_embed_kernel(
    const int* __restrict__ x, const float* __restrict__ emb,
    const float* __restrict__ Wq, const float* __restrict__ Wk,
    const float* __restrict__ Wv,
    float* __restrict__ Q, float* __restrict__ K, float* __restrict__ V) {
  const int mtile = blockIdx.x;
  const int ntile = blockIdx.y;
  const int lane  = threadIdx.x;
  const int l16   = lane & 15;
  const int hi    = lane >> 4;

  const float* W;
  float* O;
  const int col0 = (ntile & 3) * 16;
  if (ntile < 4)      { W = Wq; O = Q; }
  else if (ntile < 8) { W = Wk; O = K; }
  else                { W = Wv; O = V; }

  const int tok   = x[mtile * 16 + l16];
  const int hbase = tok * E_;
  const int wbase = (col0 + l16) * E_;

  v8f c = {};
  for (int kk = 0; kk < E_; kk += 4) {
    v2f a = *(const v2f*)(emb + hbase + kk + 2 * hi);
    v2f b = *(const v2f*)(W + wbase + kk + 2 * hi);
    c = wmma_f32(a, b, c);
  }

#pragma unroll
  for (int r = 0; r < 8; ++r) {
    const int m = mtile * 16 + r + 8 * hi;
    O[m * D_ + col0 + l16] = c[r];
  }
}

// ---------------------------------------------------------------------------
// Kernel 2: flash attention. Grid: (S/16, B); one wave per 16-query tile.
// K/V tiles DMA'd into LDS by the TDM (double-buffered on TENSORcnt).
// ---------------------------------------------------------------------------
__global__ __launch_bounds__(32) void flash_attn_kernel(
    const float* __restrict__ Q, const float* __restrict__ K,
    const float* __restrict__ V, float* __restrict__ out) {
  __shared__ float kv_lds[4 * TILEDW];    // [buf0:K,V][buf1:K,V], stride-66 rows
  __shared__ float plds[16 * 17];         // P transpose scratch

  const int qt   = blockIdx.x;
  const int b    = blockIdx.y;
  const int lane = threadIdx.x;
  const int l16  = lane & 15;
  const int hi   = lane >> 4;

  const float* Qb = Q + b * S_ * D_;
  const float* Kb = K + b * S_ * D_;
  const float* Vb = V + b * S_ * D_;

  // Preload Q A-fragments, folding in the 1/sqrt(256) score scale.
  const int qrow = qt * 16 + l16;
  v2f qf[16];
#pragma unroll
  for (int st = 0; st < 16; ++st) {
    v2f q = *(const v2f*)(Qb + qrow * D_ + st * 4 + 2 * hi);
    qf[st] = q * 0.0625f;
  }

  float mrow[8], lrow[8];
#pragma unroll
  for (int r = 0; r < 8; ++r) { mrow[r] = -INFINITY; lrow[r] = 0.0f; }
  v8f acc[4] = {{}, {}, {}, {}};

  const unsigned kv_base = (unsigned)(uintptr_t)&kv_lds[0];

  // Prologue: DMA K/V tiles for block 0 into buffer 0.
  tdm_load_tile16x64(kv_base, Kb);
  tdm_load_tile16x64(kv_base + TILEDW * 4, Vb);

  for (int kb = 0; kb < NB_; ++kb) {
    const int cur = kb & 1;
    if (kb + 1 < NB_) {
      const int nxt = cur ^ 1;
      const unsigned noff = kv_base + nxt * (2 * TILEDW * 4);
      tdm_load_tile16x64(noff, Kb + (kb + 1) * 16 * D_);
      tdm_load_tile16x64(noff + TILEDW * 4, Vb + (kb + 1) * 16 * D_);
      __builtin_amdgcn_s_wait_tensorcnt(2);   // current buffer's 2 DMAs done
    } else {
      __builtin_amdgcn_s_wait_tensorcnt(0);
    }
    asm volatile("" ::: "memory");            // order LDS reads after TDM wait

    const float* Kl = kv_lds + cur * (2 * TILEDW);
    const float* Vl = Kl + TILEDW;

    // S = Q . K^T : B[d][key] = K[key][d] read from padded LDS tile.
    v8f s = {};
#pragma unroll
    for (int st = 0; st < 16; ++st) {
      v2f bf = *(const v2f*)(Kl + l16 * LSTRIDE + st * 4 + 2 * hi);
      s = wmma_f32(qf[st], bf, s);
    }

    // Online softmax; shfl_xor masks 1/2/4/8 stay within each 16-lane half,
    // matching the D layout (lanes 0-15: rows r, lanes 16-31: rows r+8).
    float pr[8];
#pragma unroll
    for (int r = 0; r < 8; ++r) {
      float sv = s[r];
      float mx = sv;
      mx = fmaxf(mx, __shfl_xor(mx, 1));
      mx = fmaxf(mx, __shfl_xor(mx, 2));
      mx = fmaxf(mx, __shfl_xor(mx, 4));
      mx = fmaxf(mx, __shfl_xor(mx, 8));
      const float mnew  = fmaxf(mrow[r], mx);
      const float alpha = __expf(mrow[r] - mnew);
      const float p     = __expf(sv - mnew);
      float rs = p;
      rs += __shfl_xor(rs, 1);
      rs += __shfl_xor(rs, 2);
      rs += __shfl_xor(rs, 4);
      rs += __shfl_xor(rs, 8);
      lrow[r] = lrow[r] * alpha + rs;
      mrow[r] = mnew;
#pragma unroll
      for (int dc = 0; dc < 4; ++dc) acc[dc][r] *= alpha;
      pr[r] = p;
    }

    // Transpose P (D layout -> A layout) through LDS.
#pragma unroll
    for (int r = 0; r < 8; ++r) plds[(r + 8 * hi) * 17 + l16] = pr[r];
    __builtin_amdgcn_wave_barrier();
    asm volatile("s_wait_dscnt 0x0" ::: "memory");  // LDS RAW fence (same wave)

    // O += P . V : A from LDS transpose, B[key][d] from padded V tile.
#pragma unroll
    for (int st = 0; st < 4; ++st) {
      v2f af;
      af.x = plds[l16 * 17 + st * 4 + 2 * hi + 0];
      af.y = plds[l16 * 17 + st * 4 + 2 * hi + 1];
#pragma unroll
      for (int dc = 0; dc < 4; ++dc) {
        v2f bf;
        bf.x = Vl[(st * 4 + 2 * hi + 0) * LSTRIDE + dc * 16 + l16];
        bf.y = Vl[(st * 4 + 2 * hi + 1) * LSTRIDE + dc * 16 + l16];
        acc[dc] = wmma_f32(af, bf, acc[dc]);
      }
    }
    __builtin_amdgcn_wave_barrier();
  }

  // Normalize and store [B,S,64].
  float* Ob = out + b * S_ * D_;
#pragma unroll
  for (int r = 0; r < 8; ++r) {
    const float inv = 1.0f / lrow[r];
    const int m = qt * 16 + r + 8 * hi;
#pragma unroll
    for (int dc = 0; dc < 4; ++dc)
      Ob[m * D_ + dc * 16 + l16] = acc[dc][r] * inv;
  }
}

// ---------------------------------------------------------------------------
extern "C" void kernel_launch(void* const* d_in, const int* in_sizes, int n_in,
                              void* d_out, int out_size, void* d_ws,
                              size_t ws_size, hipStream_t stream) {
  const int*   x   = (const int*)d_in[0];
  const float* emb = (const float*)d_in[1];
  const float* Wq  = (const float*)d_in[2];
  const float* Wk  = (const float*)d_in[3];
  const float* Wv  = (const float*)d_in[4];
  float* out = (float*)d_out;

  float* Qw = (float*)d_ws;            // [B*S, 64]
  float* Kw = Qw + B_ * S_ * D_;       // [B*S, 64]
  float* Vw = Kw + B_ * S_ * D_;       // [B*S, 64]  (12 MB total in d_ws)

  qkv_embed_kernel<<<dim3((B_ * S_) / 16, 12), 32, 0, stream>>>(
      x, emb, Wq, Wk, Wv, Qw, Kw, Vw);
  flash_attn_kernel<<<dim3(S_ / 16, B_), 32, 0, stream>>>(Qw, Kw, Vw, out);
}